// PosEncoding_fix_2207613190388
// MI455X (gfx1250) — compile-verified
//
#include <hip/hip_runtime.h>

// Positional-encoding "fix" kernel for MI455X (gfx1250).
// out[row, i] = (i even) ? sin(pos[row]*w_k[i]) : cos(pos[row]*w_k[i]);  row zeroed if pos==0.
// Memory-bound: 512MB streaming output -> NT b128 stores; positions staged via
// gfx1250 async global->LDS loads (ASYNCcnt); HW sin/cos with manual range reduction.

typedef float v4f __attribute__((ext_vector_type(4)));

#define TILE_ROWS 256
#define BLOCK     256

__global__ __launch_bounds__(BLOCK) void pos_enc_fix_kernel(
    const float* __restrict__ positions,
    const float* __restrict__ w_k,
    float* __restrict__ out,
    int n)
{
    __shared__ float s_pos[TILE_ROWS];

    const int tid       = threadIdx.x;
    const int lane      = tid & 31;
    const int wave      = tid >> 5;
    const int tile_base = blockIdx.x * TILE_ROWS;

    // ---- Stage this block's 256 positions into LDS. Each wave issues ONE
    // async vector load for its own 32 lanes, then waits on ASYNCcnt.
    int idx = tile_base + tid;
    if (idx > n - 1) idx = n - 1;   // clamp tail (harmless re-read)

#if __has_builtin(__builtin_amdgcn_global_load_async_to_lds_b32)
    __builtin_amdgcn_global_load_async_to_lds_b32(
        (__attribute__((address_space(1))) int*)(positions + idx),
        (__attribute__((address_space(3))) int*)&s_pos[tid],
        /*offset=*/0, /*cpol=*/0);
#if __has_builtin(__builtin_amdgcn_s_wait_asynccnt)
    __builtin_amdgcn_s_wait_asynccnt(0);
#else
    asm volatile("s_wait_asynccnt 0" ::: "memory");
#endif
    asm volatile("" ::: "memory");   // keep LDS reads below the wait
#else
    // Fallback: plain LDS store (compiler tracks DScnt dependence).
    s_pos[tid] = positions[idx];
#endif

    // ---- Per-lane frequencies: channels [4*lane, 4*lane+4).
    // w_k[2j] == w_k[2j+1], so pairs (c,c+1) and (c+2,c+3) use wv.x and wv.z.
    const int c = lane * 4;
    const v4f wv = *(const v4f*)(w_k + c);       // 16B-aligned b128 load
    const float wA = wv.x;
    const float wB = wv.z;
    const float inv2pi = 0.15915494309189535f;   // 1/(2*pi)

    const int wave_row0 = tile_base + wave * 32;
    float* outp = out + (size_t)wave_row0 * 128 + c;

    // ---- 32 rows per wave; each lane emits one float4 (sin,cos,sin,cos).
#pragma unroll 4
    for (int r = 0; r < 32; ++r) {
        const int row = wave_row0 + r;
        if (row >= n) break;
        const float p = s_pos[wave * 32 + r];    // LDS broadcast (same addr all lanes)

        // Range-reduce in revolutions: v_sin_f32/v_cos_f32 take input in
        // units of revolutions with limited domain; fract puts it in [0,1).
        const float rA = __builtin_amdgcn_fractf(p * wA * inv2pi);
        const float rB = __builtin_amdgcn_fractf(p * wB * inv2pi);

        v4f v;
        v.x = __builtin_amdgcn_sinf(rA);
        v.y = __builtin_amdgcn_cosf(rA);
        v.z = __builtin_amdgcn_sinf(rB);
        v.w = __builtin_amdgcn_cosf(rB);

        if (p == 0.0f) {                         // "fix": zero embedding at pos==0
            v.x = 0.0f; v.y = 0.0f; v.z = 0.0f; v.w = 0.0f;
        }

        // 512MB stream >> 192MB L2: bypass with non-temporal b128 store.
        __builtin_nontemporal_store(v, (v4f*)(outp + (size_t)r * 128));
    }
}

extern "C" void kernel_launch(void* const* d_in, const int* in_sizes, int n_in,
                              void* d_out, int out_size, void* d_ws, size_t ws_size,
                              hipStream_t stream) {
    const float* positions = (const float*)d_in[0];
    const float* w_k       = (const float*)d_in[1];
    float*       out       = (float*)d_out;
    const int n = in_sizes[0];

    const int blocks = (n + TILE_ROWS - 1) / TILE_ROWS;  // 3907 for n=1e6
    pos_enc_fix_kernel<<<blocks, BLOCK, 0, stream>>>(positions, w_k, out, n);
}